// FullyConnectedLayer_3006477107724
// MI455X (gfx1250) — compile-verified
//
#include <hip/hip_runtime.h>
#include <math.h>

#define B_ROWS 4096
#define K_DIM  512
#define N_COLS 10575

typedef float v2f __attribute__((ext_vector_type(2)));
typedef float v8f __attribute__((ext_vector_type(8)));

#define WMMA_F32(A_, B_, C_) \
    __builtin_amdgcn_wmma_f32_16x16x4_f32(false, (A_), false, (B_), (short)0, (C_), false, false)

// ---------------------------------------------------------------------------
// Row L2-normalization: one 256-thread block per 512-element row.
// ---------------------------------------------------------------------------
__global__ void __launch_bounds__(256) norm_rows_k(const float* __restrict__ in,
                                                   float* __restrict__ out)
{
    const int row = blockIdx.x;
    const int t   = threadIdx.x;
    const float* src = in  + (size_t)row * K_DIM;
    float*       dst = out + (size_t)row * K_DIM;

    float v0 = src[t];
    float v1 = src[t + 256];

    __shared__ float red[256];
    red[t] = v0 * v0 + v1 * v1;
    __syncthreads();
    for (int s = 128; s > 0; s >>= 1) {
        if (t < s) red[t] += red[t + s];
        __syncthreads();
    }
    const float scale = 1.0f / fmaxf(sqrtf(red[0]), 1e-12f);
    dst[t]       = v0 * scale;
    dst[t + 256] = v1 * scale;
}

// ---------------------------------------------------------------------------
// cos_theta GEMM on the f32 WMMA path.
// Block = 4 waves in a 2x2 arrangement -> 64(M) x 128(N) block tile.
// Each wave owns 32(M) x 64(N): 2 A-frags x 4 B-frags -> 8 independent
// accumulator chains, 8 V_WMMA_F32_16X16X4_F32 per K-step for 6 b64 loads.
// Fragments are register double-buffered (prefetch next K-step before the
// WMMAs) so loadcnt waits overlap matrix work. Clip + *32 fused into store.
// ---------------------------------------------------------------------------
__global__ void __launch_bounds__(128) cos_gemm_wmma(const float* __restrict__ A,
                                                     const float* __restrict__ W,
                                                     float* __restrict__ out)
{
    const int lane = threadIdx.x & 31;
    const int wave = threadIdx.x >> 5;      // 0..3
    const int wy   = wave >> 1;             // 0..1  (M)
    const int wx   = wave & 1;              // 0..1  (N)
    const int lid  = lane & 15;             // position within half-wave
    const int half = lane >> 4;             // 0: K pair {0,1}; 1: K pair {2,3}

    const int mWave = blockIdx.y * 64 + wy * 32;     // 4096/64 = 64 exact
    const int nWave = blockIdx.x * 128 + wx * 64;    // 83*128 covers 10575

    // A fragment pointers: lane -> row, contiguous K pair at (k + 2*half).
    const float* aPtr0 = A + (size_t)(mWave + lid)      * K_DIM + half * 2;
    const float* aPtr1 = A + (size_t)(mWave + 16 + lid) * K_DIM + half * 2;

    // B fragment pointers: B[k][n] = Wn[n][k]; lane -> column, contiguous K pair.
    const int c0 = nWave + lid;
    const int c1 = c0 + 16;
    const int c2 = c0 + 32;
    const int c3 = c0 + 48;
    const int cc0 = (c0 < N_COLS) ? c0 : (N_COLS - 1);
    const int cc1 = (c1 < N_COLS) ? c1 : (N_COLS - 1);
    const int cc2 = (c2 < N_COLS) ? c2 : (N_COLS - 1);
    const int cc3 = (c3 < N_COLS) ? c3 : (N_COLS - 1);
    const float* b0p = W + (size_t)cc0 * K_DIM + half * 2;
    const float* b1p = W + (size_t)cc1 * K_DIM + half * 2;
    const float* b2p = W + (size_t)cc2 * K_DIM + half * 2;
    const float* b3p = W + (size_t)cc3 * K_DIM + half * 2;

    v8f acc00 = {}, acc01 = {}, acc02 = {}, acc03 = {};
    v8f acc10 = {}, acc11 = {}, acc12 = {}, acc13 = {};

    // Prologue: load K-step 0 fragments.
    v2f a0c = *(const v2f*)(aPtr0);
    v2f a1c = *(const v2f*)(aPtr1);
    v2f b0c = *(const v2f*)(b0p);
    v2f b1c = *(const v2f*)(b1p);
    v2f b2c = *(const v2f*)(b2p);
    v2f b3c = *(const v2f*)(b3p);

#pragma unroll 2
    for (int k = 4; k < K_DIM; k += 4) {
        // Prefetch next K-step fragments (in flight during the 8 WMMAs below).
        v2f a0n = *(const v2f*)(aPtr0 + k);
        v2f a1n = *(const v2f*)(aPtr1 + k);
        v2f b0n = *(const v2f*)(b0p + k);
        v2f b1n = *(const v2f*)(b1p + k);
        v2f b2n = *(const v2f*)(b2p + k);
        v2f b3n = *(const v2f*)(b3p + k);

        acc00 = WMMA_F32(a0c, b0c, acc00);
        acc10 = WMMA_F32(a1c, b0c, acc10);
        acc01 = WMMA_F32(a0c, b1c, acc01);
        acc11 = WMMA_F32(a1c, b1c, acc11);
        acc02 = WMMA_F32(a0c, b2c, acc02);
        acc12 = WMMA_F32(a1c, b2c, acc12);
        acc03 = WMMA_F32(a0c, b3c, acc03);
        acc13 = WMMA_F32(a1c, b3c, acc13);

        a0c = a0n; a1c = a1n;
        b0c = b0n; b1c = b1n; b2c = b2n; b3c = b3n;
    }

    // Epilogue K-step.
    acc00 = WMMA_F32(a0c, b0c, acc00);
    acc10 = WMMA_F32(a1c, b0c, acc10);
    acc01 = WMMA_F32(a0c, b1c, acc01);
    acc11 = WMMA_F32(a1c, b1c, acc11);
    acc02 = WMMA_F32(a0c, b2c, acc02);
    acc12 = WMMA_F32(a1c, b2c, acc12);
    acc03 = WMMA_F32(a0c, b3c, acc03);
    acc13 = WMMA_F32(a1c, b3c, acc13);

    // C/D layout: VGPR r, lanes0-15 -> M=r, lanes16-31 -> M=8+r; N = lid.
    const int row0 = mWave + half * 8;        // m-tile 0
    const int row1 = mWave + 16 + half * 8;   // m-tile 1
#pragma unroll
    for (int r = 0; r < 8; ++r) {
        float* orow0 = out + (size_t)(row0 + r) * N_COLS;
        float* orow1 = out + (size_t)(row1 + r) * N_COLS;
        float v;
        v = fminf(fmaxf(acc00[r], -1.0f), 1.0f) * 32.0f; if (c0 < N_COLS) orow0[c0] = v;
        v = fminf(fmaxf(acc01[r], -1.0f), 1.0f) * 32.0f; if (c1 < N_COLS) orow0[c1] = v;
        v = fminf(fmaxf(acc02[r], -1.0f), 1.0f) * 32.0f; if (c2 < N_COLS) orow0[c2] = v;
        v = fminf(fmaxf(acc03[r], -1.0f), 1.0f) * 32.0f; if (c3 < N_COLS) orow0[c3] = v;
        v = fminf(fmaxf(acc10[r], -1.0f), 1.0f) * 32.0f; if (c0 < N_COLS) orow1[c0] = v;
        v = fminf(fmaxf(acc11[r], -1.0f), 1.0f) * 32.0f; if (c1 < N_COLS) orow1[c1] = v;
        v = fminf(fmaxf(acc12[r], -1.0f), 1.0f) * 32.0f; if (c2 < N_COLS) orow1[c2] = v;
        v = fminf(fmaxf(acc13[r], -1.0f), 1.0f) * 32.0f; if (c3 < N_COLS) orow1[c3] = v;
    }
}

// ---------------------------------------------------------------------------
// Per-row rival selection + margin scatter. out already holds clip(cos)*32.
// ---------------------------------------------------------------------------
__global__ void __launch_bounds__(256) row_finalize(float* __restrict__ out,
                                                    const int* __restrict__ label)
{
    const int row = blockIdx.x;
    const int t   = threadIdx.x;
    float* orow = out + (size_t)row * N_COLS;
    const float inv = 1.0f / 32.0f;
    const int NONE = 0x7FFFFFFF;

    // Pass 1: max cos (first-occurrence argmax) + unstabilized sum of exp
    // (cos in [-1,1] so exp is well-conditioned in f32).
    float bestV = -2.0f; int bestI = NONE; float sumE = 0.0f;
    for (int j = t; j < N_COLS; j += 256) {
        const float c = orow[j] * inv;
        sumE += expf(c);
        if (c > bestV) { bestV = c; bestI = j; }
    }

    __shared__ float sv[256];
    __shared__ int   si[256];
    __shared__ float ssum[256];
    sv[t] = bestV; si[t] = bestI; ssum[t] = sumE;
    __syncthreads();
    for (int s = 128; s > 0; s >>= 1) {
        if (t < s) {
            ssum[t] += ssum[t + s];
            const float v2 = sv[t + s]; const int i2 = si[t + s];
            if (v2 > sv[t] || (v2 == sv[t] && i2 < si[t])) { sv[t] = v2; si[t] = i2; }
        }
        __syncthreads();
    }
    const float m      = sv[0];
    const int   maxIdx = si[0];
    const float S      = ssum[0];
    __syncthreads();

    // soft_j < max_num - 1e-4  <=>  exp(c_j) < exp(m) - 1e-4 * S
    const float lim = expf(m) - 1.0e-4f * S;

    // Pass 2: first-occurrence argmax among qualifying columns.
    float bV = -3.0f; int bI = NONE;
    for (int j = t; j < N_COLS; j += 256) {
        const float c = orow[j] * inv;
        if (expf(c) < lim) {
            if (c > bV) { bV = c; bI = j; }
        }
    }
    sv[t] = bV; si[t] = bI;
    __syncthreads();
    for (int s = 128; s > 0; s >>= 1) {
        if (t < s) {
            const float v2 = sv[t + s]; const int i2 = si[t + s];
            if (v2 > sv[t] || (v2 == sv[t] && i2 < si[t])) { sv[t] = v2; si[t] = i2; }
        }
        __syncthreads();
    }

    if (t == 0) {
        const int lbl    = label[row];
        const int second = (si[0] == NONE) ? 0 : si[0];   // all-masked -> argmax(0s)=0
        const int rival  = (maxIdx == lbl) ? second : maxIdx;

        const float cosin    = orow[lbl]   * inv;   // already clipped
        const float rivalCos = orow[rival] * inv;

        const float score  = (cosin > 0.0f)    ? (cosin - 0.2f)              : cosin;
        const float rscore = (rivalCos > 0.6f) ? ((rivalCos + 0.2f) * 0.2f)  : rivalCos;

        orow[rival] = rscore * 32.0f;   // rival first,
        orow[lbl]   = score  * 32.0f;   // label overwrites on collision
    }
}

// ---------------------------------------------------------------------------
extern "C" void kernel_launch(void* const* d_in, const int* in_sizes, int n_in,
                              void* d_out, int out_size, void* d_ws, size_t ws_size,
                              hipStream_t stream)
{
    const float* x     = (const float*)d_in[0];   // [4096, 512]
    const int*   label = (const int*)d_in[1];     // [4096]
    const float* w     = (const float*)d_in[2];   // [10575, 512]
    float*       out   = (float*)d_out;           // [4096, 10575]

    float* xn = (float*)d_ws;                     // 4096*512 floats  (8 MB)
    float* wn = xn + (size_t)B_ROWS * K_DIM;      // 10575*512 floats (21.7 MB)

    norm_rows_k<<<B_ROWS, 256, 0, stream>>>(x, xn);
    norm_rows_k<<<N_COLS, 256, 0, stream>>>(w, wn);

    dim3 grid((N_COLS + 127) / 128, B_ROWS / 64); // 83 x 64 blocks, 128 thr
    cos_gemm_wmma<<<grid, 128, 0, stream>>>(xn, wn, out);

    row_finalize<<<B_ROWS, 256, 0, stream>>>(out, label);
}